// kernel_generated_1_21887153341342
// MI455X (gfx1250) — compile-verified
//
#include <hip/hip_runtime.h>

// CDNA5 (gfx1250) wave32 WMMA types
typedef float v2f __attribute__((ext_vector_type(2)));
typedef float v4f __attribute__((ext_vector_type(4)));
typedef float v8f __attribute__((ext_vector_type(8)));

// Reference collapses to a cyclic 2x3 conv:
//   y[i,o,n] = sum_{j,k} x[j,(o-1)%28,n+k-1]*W[i,j,0,k] + x[j,(o-2)%28,n+k-1]*W[i,j,1,k]
// GEMM view: M=784 spatial (49x16 tiles), N=128 channels (8x16 tiles),
//            K=384 ordered K = j*6 + t  (j = input channel, t = l*3+k tap index).
// With this ordering B[K][i] == W_flat[i*384 + K]: the B operand is a contiguous
// row of W and every per-lane B fetch is an aligned immediate-offset b64 load.
#define K_TOT   384
#define A_LDSTR 388   // 384 + 4 pad: row stride not a multiple of the 64 LDS banks

__global__ __launch_bounds__(256) void conv_roll_wmma_f32(
    const float* __restrict__ x,   // 64*28*28
    const float* __restrict__ W,   // 128*64*2*3 (i,j,l,k)
    float* __restrict__ out)       // 128*28*28
{
    __shared__ float As[16 * A_LDSTR];   // 16 spatial rows x K=384 im2col tile (~24.8 KB)

    const int tid   = threadIdx.x;
    const int mtile = blockIdx.x;        // 49 tiles of 16 spatial positions

    // ---- Cooperative im2col into LDS.
    // Thread owns spatial row mm = tid&15 and channels j0..j0+3 (j0 = (tid>>4)*4),
    // i.e. K-range [(tid>>4)*24, +24). One small division per thread; the
    // per-element index math is pure shifts/adds (no div by 6 needed).
    {
        const int mm = tid & 15;
        const int j0 = (tid >> 4) << 2;
        const int m  = mtile * 16 + mm;
        const int o  = m / 28;
        const int n  = m - o * 28;
        #pragma unroll
        for (int jj = 0; jj < 4; ++jj) {
            const int j = j0 + jj;
            #pragma unroll
            for (int t = 0; t < 6; ++t) {
                int l   = (t >= 3) ? 1 : 0;
                int k   = t - l * 3;
                int row = o - 1 - l; if (row < 0) row += 28;   // cyclic row wrap
                int col = n + k - 1;                           // zero-padded column
                float v = (col >= 0 && col < 28) ? x[j * 784 + row * 28 + col] : 0.0f;
                As[mm * A_LDSTR + j * 6 + t] = v;
            }
        }
    }
    __syncthreads();

    // ---- Per-wave 16x16 output tile: wave w owns channels [w*16, w*16+16) ----
    const int lane  = tid & 31;
    const int wv    = tid >> 5;              // 0..7 = N tile
    const int mloc  = lane & 15;             // M (A) / N (B,D) position within tile
    const int khalf = (lane >> 4) << 1;      // lanes 16-31 carry K+2,K+3

    // Fixed base pointers; every K-step below is an immediate offset off these.
    const float* aPtr = &As[mloc * A_LDSTR + khalf];    // A[mloc][4*kk+khalf .. +1]
    const int    ich  = wv * 16 + mloc;
    const float* wPtr = W + ich * K_TOT + khalf;        // B[K][ich] = W_flat[ich*384 + K]

    v8f c = {0.f, 0.f, 0.f, 0.f, 0.f, 0.f, 0.f, 0.f};

    // Fully unrolled 96-step K loop: one ds_load_b64 (A) + one global_load_b64 (B)
    // + one v_wmma_f32_16x16x4_f32 per step; all addresses are immediates.
    #pragma unroll
    for (int kk = 0; kk < 96; ++kk) {
        const int kbase = kk * 4;                      // compile-time constant
        v2f a = *(const v2f*)(aPtr + kbase);           // 8B-aligned LDS read
        v2f b = *(const v2f*)(wPtr + kbase);           // 8B-aligned global read
        c = __builtin_amdgcn_wmma_f32_16x16x4_f32(
                false, a, false, b, (short)0, c, false, false);
    }

    // ---- Store D: VGPR r holds M = r (lanes 0-15) / r+8 (lanes 16-31), N = mloc.
    // Flattened y index = i*784 + m is contiguous in r -> two 16B stores.
    const int mbase = mtile * 16 + ((lane >> 4) << 3);
    float* po = out + ich * 784 + mbase;
    v4f lo = {c[0], c[1], c[2], c[3]};
    v4f hi = {c[4], c[5], c[6], c[7]};
    *(v4f*)(po)     = lo;
    *(v4f*)(po + 4) = hi;
}

extern "C" void kernel_launch(void* const* d_in, const int* in_sizes, int n_in,
                              void* d_out, int out_size, void* d_ws, size_t ws_size,
                              hipStream_t stream) {
    (void)in_sizes; (void)n_in; (void)d_ws; (void)ws_size; (void)out_size;
    const float* x = (const float*)d_in[0];   // (1,64,28,28)
    const float* W = (const float*)d_in[1];   // (128,64,2,3)
    float* out     = (float*)d_out;           // (1,128,28,28)
    conv_roll_wmma_f32<<<49, 256, 0, stream>>>(x, W, out);
}